// MultiHead_Covar_Attn_80066780332412
// MI455X (gfx1250) — compile-verified
//
#include <hip/hip_runtime.h>

typedef __attribute__((ext_vector_type(2))) float v2f;
typedef __attribute__((ext_vector_type(8))) float v8f;
typedef int vint4 __attribute__((vector_size(16)));

#define BB 32
#define CC 128
#define MM 4000
#define TILE_M 128
#define NT 32          // ceil(4000/128)
#define LDP 132        // padded LDS row stride (floats)
#define GROUPS 4
#define DGRP 32

static_assert((MM + TILE_M - 1) / TILE_M == NT, "tile count");

__device__ __forceinline__ v8f wmma_f32(v2f a, v2f b, v8f c) {
  // D(16x16,f32) = A(16x4,f32) * B(4x16,f32) + C
  return __builtin_amdgcn_wmma_f32_16x16x4_f32(false, a, false, b, (short)0, c,
                                               false, false);
}

// ---- CDNA5 async global->LDS copy (ASYNCcnt-tracked), guarded -------------
#if defined(__has_builtin)
#if __has_builtin(__builtin_amdgcn_global_load_async_to_lds_b128)
#define HAVE_ASYNC_LDS 1
#endif
#endif
#ifndef HAVE_ASYNC_LDS
#define HAVE_ASYNC_LDS 0
#endif

#if HAVE_ASYNC_LDS
__device__ __forceinline__ void async_copy16(const float* g, float* l) {
  // prototype: (v4i addrspace(1)* src, v4i addrspace(3)* dst, imm off, imm cpol)
  __builtin_amdgcn_global_load_async_to_lds_b128(
      (__attribute__((address_space(1))) vint4*)(g),
      (__attribute__((address_space(3))) vint4*)(l), 0, 0);
}
__device__ __forceinline__ void wait_async() {
#if __has_builtin(__builtin_amdgcn_s_wait_asynccnt)
  __builtin_amdgcn_s_wait_asynccnt(0);
#else
  asm volatile("s_wait_asynccnt 0x0" ::: "memory");
#endif
}
#endif

// ---------------------------------------------------------------------------
// Fused 128x128 @ 128x4000 GEMM (per batch), WMMA f32 16x16x4.
// MODE 1: feat = Wl @ x;            epilogue: per-tile block-diag Gram + rowsums
// MODE 2: h1 = W1 @ relu(a*y+b)+b1; epilogue: per-tile per-channel sum/sumsq
// MODE 3: out = W2 @ relu(a*h1+b)+b2 + y (residual)
// ---------------------------------------------------------------------------
template <int MODE>
__global__ __launch_bounds__(256) void gemm128(
    const float* __restrict__ src, const float* __restrict__ Wm,
    const float* __restrict__ bias, const float* __restrict__ scl,
    const float* __restrict__ sft, float* __restrict__ dst,
    const float* __restrict__ resid, float* __restrict__ gram_p,
    float* __restrict__ rsum_p, float* __restrict__ st_sum,
    float* __restrict__ st_sq) {
  __shared__ __align__(16) float Wl[CC * LDP];
  __shared__ __align__(16) float Xl[CC * LDP];

  const int b = blockIdx.y;
  const int tile = blockIdx.x;
  const int m0 = tile * TILE_M;
  const int tid = threadIdx.x;
  const int lane = tid & 31;
  const int wv = tid >> 5;
  const int half = lane >> 4;
  const int lid = lane & 15;
  const int o0 = wv * 16;

  bool need_manual_x = true;

#if HAVE_ASYNC_LDS
  // weights (128x128) stream straight into LDS; overlaps with X-tile fetch
  for (int i = tid; i < CC * 32; i += 256) {
    const int r = i >> 5, c4 = (i & 31) << 2;
    async_copy16(Wm + r * CC + c4, &Wl[r * LDP + c4]);
  }
  if (MODE == 1 && m0 + TILE_M <= MM) {  // raw full tile: direct async copy
    for (int i = tid; i < CC * 32; i += 256) {
      const int r = i >> 5, c4 = (i & 31) << 2;
      async_copy16(src + ((size_t)b * CC + r) * MM + m0 + c4,
                   &Xl[r * LDP + c4]);
    }
    need_manual_x = false;
  }
#else
  for (int i = tid; i < CC * 32; i += 256) {
    const int r = i >> 5, c4 = (i & 31) << 2;
    *(float4*)&Wl[r * LDP + c4] = *(const float4*)(Wm + r * CC + c4);
  }
#endif

  // --- src tile into LDS via VGPRs (fused affine+relu for MODE>=2, pad) ---
  if (need_manual_x) {
    for (int i = tid; i < CC * 32; i += 256) {
      const int r = i >> 5, c4 = (i & 31) << 2;
      const int m = m0 + c4;
      const size_t base = ((size_t)b * CC + r) * MM;
      float4 v;
      if (m + 3 < MM) {
        v = *(const float4*)(src + base + m);
      } else {
        v.x = (m + 0 < MM) ? src[base + m + 0] : 0.f;
        v.y = (m + 1 < MM) ? src[base + m + 1] : 0.f;
        v.z = (m + 2 < MM) ? src[base + m + 2] : 0.f;
        v.w = (m + 3 < MM) ? src[base + m + 3] : 0.f;
      }
      if (MODE >= 2) {
        const float sc = scl[b * CC + r], sh = sft[b * CC + r];
        v.x = (m + 0 < MM) ? fmaxf(fmaf(sc, v.x, sh), 0.f) : 0.f;
        v.y = (m + 1 < MM) ? fmaxf(fmaf(sc, v.y, sh), 0.f) : 0.f;
        v.z = (m + 2 < MM) ? fmaxf(fmaf(sc, v.z, sh), 0.f) : 0.f;
        v.w = (m + 3 < MM) ? fmaxf(fmaf(sc, v.w, sh), 0.f) : 0.f;
      }
      *(float4*)&Xl[r * LDP + c4] = v;
    }
  }

#if HAVE_ASYNC_LDS
  wait_async();  // this wave's async LDS writes landed
#endif
  __syncthreads();

  const v8f vz = {0.f, 0.f, 0.f, 0.f, 0.f, 0.f, 0.f, 0.f};
  v8f acc[8];
#pragma unroll
  for (int mt = 0; mt < 8; ++mt) acc[mt] = vz;

  // K loop: 128 / 4 = 32 WMMA steps; wave computes rows [o0,o0+16) x 128 cols
  for (int kk = 0; kk < 32; ++kk) {
    const int kb = kk * 4 + 2 * half;
    const v2f a = *(const v2f*)&Wl[(o0 + lid) * LDP + kb];  // 8B aligned
#pragma unroll
    for (int mt = 0; mt < 8; ++mt) {
      v2f bf;
      bf.x = Xl[kb * LDP + mt * 16 + lid];
      bf.y = Xl[(kb + 1) * LDP + mt * 16 + lid];
      acc[mt] = wmma_f32(a, bf, acc[mt]);
    }
  }

  // --- epilogue: bias / residual / store ---
#pragma unroll
  for (int mt = 0; mt < 8; ++mt) {
    const int m = m0 + mt * 16 + lid;
#pragma unroll
    for (int r = 0; r < 8; ++r) {
      const int o = o0 + r + 8 * half;
      float v = acc[mt][r];
      if (MODE >= 2) v += bias[o];
      if (m < MM) {
        if (MODE == 3) v += resid[((size_t)b * CC + o) * MM + m];
        dst[((size_t)b * CC + o) * MM + m] = v;
      }
      acc[mt][r] = v;  // keep (with bias) for statistics epilogues
    }
  }

  if (MODE == 1) {
    // stash feat tile to LDS (padded cols are exact zeros: Wl @ 0)
    __syncthreads();
#pragma unroll
    for (int mt = 0; mt < 8; ++mt)
#pragma unroll
      for (int r = 0; r < 8; ++r)
        Xl[(o0 + r + 8 * half) * LDP + mt * 16 + lid] = acc[mt][r];
    __syncthreads();
    // per-channel row sums of feat over this tile (deterministic partials)
    if (tid < CC) {
      const int valid = (TILE_M < MM - m0) ? TILE_M : (MM - m0);
      float s = 0.f;
      for (int m = 0; m < valid; ++m) s += Xl[tid * LDP + m];
      rsum_p[((size_t)b * NT + tile) * CC + tid] = s;
    }
    // block-diagonal Gram: group g = wv>>1, contract over 128 local cols (m)
    const int g = wv >> 1;
    v8f ga[2];
    ga[0] = vz;
    ga[1] = vz;
    for (int kk = 0; kk < 32; ++kk) {
      const int kb = kk * 4 + 2 * half;
      const v2f a = *(const v2f*)&Xl[(o0 + lid) * LDP + kb];
#pragma unroll
      for (int nt = 0; nt < 2; ++nt) {
        const int n0 = g * DGRP + nt * 16;
        const v2f bf = *(const v2f*)&Xl[(n0 + lid) * LDP + kb];
        ga[nt] = wmma_f32(a, bf, ga[nt]);
      }
    }
    const int ig0 = (wv & 1) * 16;
    float* gslice =
        gram_p + (((size_t)b * NT + tile) * GROUPS + g) * (DGRP * DGRP);
#pragma unroll
    for (int nt = 0; nt < 2; ++nt)
#pragma unroll
      for (int r = 0; r < 8; ++r)
        gslice[(ig0 + r + 8 * half) * DGRP + nt * 16 + lid] = ga[nt][r];
  }

  if (MODE == 2) {
    __syncthreads();
#pragma unroll
    for (int mt = 0; mt < 8; ++mt)
#pragma unroll
      for (int r = 0; r < 8; ++r)
        Xl[(o0 + r + 8 * half) * LDP + mt * 16 + lid] = acc[mt][r];
    __syncthreads();
    if (tid < CC) {
      const int valid = (TILE_M < MM - m0) ? TILE_M : (MM - m0);
      float s = 0.f, q = 0.f;
      for (int m = 0; m < valid; ++m) {
        const float v = Xl[tid * LDP + m];
        s += v;
        q += v * v;
      }
      st_sum[((size_t)b * NT + tile) * CC + tid] = s;
      st_sq[((size_t)b * NT + tile) * CC + tid] = q;
    }
  }
}

// ---------------------------------------------------------------------------
// Newton-Schulz matrix sqrt on 32x32 per (b,g); builds cov from Gram partials.
// ---------------------------------------------------------------------------
__device__ __forceinline__ void mm32(float (*D)[33], const float (*A)[33],
                                     const float (*Bm)[33], int tid) {
  __syncthreads();
  for (int e = tid; e < 1024; e += 256) {
    const int i = e >> 5, j = e & 31;
    float s = 0.f;
#pragma unroll 8
    for (int k = 0; k < 32; ++k) s = fmaf(A[i][k], Bm[k][j], s);
    D[i][j] = s;
  }
  __syncthreads();
}

__global__ __launch_bounds__(256) void ns_sqrtm_kernel(
    const float* __restrict__ gram_p, const float* __restrict__ rsum_p,
    float* __restrict__ s_out) {
  __shared__ float Sb[6][32][33];
  __shared__ float rsh[32];
  __shared__ float trS;
  const int bg = blockIdx.x;
  const int b = bg >> 2, g = bg & 3;
  const int tid = threadIdx.x;
  const float invM = 1.0f / (float)MM;

  if (tid < 32) {
    float r = 0.f;
    for (int t2 = 0; t2 < NT; ++t2)
      r += rsum_p[((size_t)b * NT + t2) * CC + g * DGRP + tid];
    rsh[tid] = r;
  }
  __syncthreads();
  for (int e = tid; e < 1024; e += 256) {
    const int i = e >> 5, j = e & 31;
    float gij = 0.f;
    for (int t2 = 0; t2 < NT; ++t2)
      gij += gram_p[(((size_t)b * NT + t2) * GROUPS + g) * (DGRP * DGRP) +
                    i * 32 + j];
    Sb[0][i][j] = (gij - rsh[i] * rsh[j] * invM) * invM;  // centered cov
  }
  __syncthreads();
  if (tid == 0) {
    float t = 0.f;
    for (int i = 0; i < 32; ++i) t += Sb[0][i][i];
    trS = t;
  }
  __syncthreads();
  const float trA = trS;
  const float invTr = 1.0f / trA;
  // An = cov/tr in Sb[0]; Z = ZY0 = 0.5*(3I - An) in Sb[2]
  for (int e = tid; e < 1024; e += 256) {
    const int i = e >> 5, j = e & 31;
    const float an = Sb[0][i][j] * invTr;
    Sb[0][i][j] = an;
    Sb[2][i][j] = 0.5f * ((i == j ? 3.0f : 0.0f) - an);
  }
  float (*Y)[33] = Sb[1];
  float (*Z)[33] = Sb[2];
  float (*F1)[33] = Sb[3];
  float (*F2)[33] = Sb[4];
  float (*F3)[33] = Sb[5];
  mm32(Y, Sb[0], Z, tid);  // Y = An @ ZY0
  for (int it = 0; it < 3; ++it) {
    float (*Q)[33] = F1;
    float (*Yn)[33] = F2;
    float (*Zn)[33] = F3;
    mm32(Q, Z, Y, tid);  // P = Z @ Y
    for (int e = tid; e < 1024; e += 256) {
      const int i = e >> 5, j = e & 31;
      Q[i][j] = 0.5f * ((i == j ? 3.0f : 0.0f) - Q[i][j]);
    }
    __syncthreads();
    mm32(Yn, Y, Q, tid);  // Y' = Y @ Q
    mm32(Zn, Q, Z, tid);  // Z' = Q @ Z
    F1 = Y;
    F2 = Z;
    F3 = Q;
    Y = Yn;
    Z = Zn;
  }
  float (*Q)[33] = F1;
  mm32(Q, Z, Y, tid);
  for (int e = tid; e < 1024; e += 256) {
    const int i = e >> 5, j = e & 31;
    Q[i][j] = 0.5f * ((i == j ? 3.0f : 0.0f) - Q[i][j]);
  }
  __syncthreads();
  float (*ZY)[33] = F2;
  mm32(ZY, Y, Q, tid);  // final sqrtm / sqrt(trA)
  const float srt = sqrtf(trA);
  if (tid < 32) {
    float s = 0.f;
    for (int i = 0; i < 32; ++i) s += ZY[i][tid];  // mean over rows
    s_out[b * CC + g * DGRP + tid] = s * (1.0f / 32.0f) * srt;
  }
}

// ---------------------------------------------------------------------------
// t = s*feat + x (in place over feat); per-(b,c) sum/sumsq for group-norm.
// ---------------------------------------------------------------------------
__global__ __launch_bounds__(256) void t_gn_kernel(
    const float* __restrict__ x, float* __restrict__ feat,
    const float* __restrict__ s, float* __restrict__ tsum,
    float* __restrict__ tsq) {
  const int bc = blockIdx.x;
  const size_t base = (size_t)bc * MM;
  const float sc = s[bc];
  const int tid = threadIdx.x;
  float ls = 0.f, lq = 0.f;
  for (int m = tid; m < MM; m += 256) {
    const float t = fmaf(sc, feat[base + m], x[base + m]);
    feat[base + m] = t;
    ls += t;
    lq += t * t;
  }
  __shared__ float r1[256], r2[256];
  r1[tid] = ls;
  r2[tid] = lq;
  __syncthreads();
  for (int off = 128; off > 0; off >>= 1) {
    if (tid < off) {
      r1[tid] += r1[tid + off];
      r2[tid] += r2[tid + off];
    }
    __syncthreads();
  }
  if (tid == 0) {
    tsum[bc] = r1[0];
    tsq[bc] = r2[0];
  }
}

// ---------------------------------------------------------------------------
// y = group_norm(t); per-(b,c) instance-norm sum/sumsq of y.
// ---------------------------------------------------------------------------
__global__ __launch_bounds__(256) void gn_apply_kernel(
    const float* __restrict__ t, const float* __restrict__ tsum,
    const float* __restrict__ tsq, const float* __restrict__ gamma,
    const float* __restrict__ beta, float* __restrict__ y,
    float* __restrict__ i_sum, float* __restrict__ i_sq) {
  const int bc = blockIdx.x;
  const int b = bc >> 7, c = bc & 127, g = c >> 5;
  const size_t base = (size_t)bc * MM;
  float gs = 0.f, gq = 0.f;
  for (int c2 = g * DGRP; c2 < (g + 1) * DGRP; ++c2) {
    gs += tsum[b * CC + c2];
    gq += tsq[b * CC + c2];
  }
  const float cnt = (float)DGRP * (float)MM;
  const float mu = gs / cnt;
  const float var = gq / cnt - mu * mu;
  const float rs = rsqrtf(var + 1e-5f);
  const float ga = gamma[c] * rs;
  const float be = beta[c] - mu * ga;
  const int tid = threadIdx.x;
  float ls = 0.f, lq = 0.f;
  for (int m = tid; m < MM; m += 256) {
    const float v = fmaf(ga, t[base + m], be);
    y[base + m] = v;
    ls += v;
    lq += v * v;
  }
  __shared__ float r1[256], r2[256];
  r1[tid] = ls;
  r2[tid] = lq;
  __syncthreads();
  for (int off = 128; off > 0; off >>= 1) {
    if (tid < off) {
      r1[tid] += r1[tid + off];
      r2[tid] += r2[tid + off];
    }
    __syncthreads();
  }
  if (tid == 0) {
    i_sum[bc] = r1[0];
    i_sq[bc] = r2[0];
  }
}

// ---------------------------------------------------------------------------
// Collapse BN(IN(.)) into per-(b,c) affine. IN zero-means rows => BN mean = 0,
// BN var[c] = mean_b( varI/(varI+epsI) ).
// ---------------------------------------------------------------------------
__global__ void coef_kernel(const float* __restrict__ sum_p,
                            const float* __restrict__ sq_p, int ntiles,
                            const float* __restrict__ bn_g,
                            const float* __restrict__ bn_b,
                            float* __restrict__ scl, float* __restrict__ sft) {
  const int c = threadIdx.x;
  if (c >= CC) return;
  const float invM = 1.0f / (float)MM;
  float accv = 0.f;
  for (int b = 0; b < BB; ++b) {
    float s = 0.f, q = 0.f;
    for (int t2 = 0; t2 < ntiles; ++t2) {
      s += sum_p[((size_t)b * ntiles + t2) * CC + c];
      q += sq_p[((size_t)b * ntiles + t2) * CC + c];
    }
    const float mu = s * invM;
    const float var = q * invM - mu * mu;
    accv += var / (var + 1e-3f);
  }
  const float rsB = rsqrtf(accv / (float)BB + 1e-5f);
  const float gsc = bn_g[c] * rsB;
  const float bb2 = bn_b[c];
  for (int b = 0; b < BB; ++b) {
    float s = 0.f, q = 0.f;
    for (int t2 = 0; t2 < ntiles; ++t2) {
      s += sum_p[((size_t)b * ntiles + t2) * CC + c];
      q += sq_p[((size_t)b * ntiles + t2) * CC + c];
    }
    const float mu = s * invM;
    const float var = q * invM - mu * mu;
    const float rsI = rsqrtf(var + 1e-3f);
    scl[b * CC + c] = rsI * gsc;
    sft[b * CC + c] = bb2 - mu * rsI * gsc;
  }
}

// ---------------------------------------------------------------------------
extern "C" void kernel_launch(void* const* d_in, const int* in_sizes, int n_in,
                              void* d_out, int out_size, void* d_ws,
                              size_t ws_size, hipStream_t stream) {
  const float* x = (const float*)d_in[0];
  const float* wl = (const float*)d_in[1];
  const float* gng = (const float*)d_in[2];
  const float* gnb = (const float*)d_in[3];
  const float* bn1g = (const float*)d_in[4];
  const float* bn1b = (const float*)d_in[5];
  const float* w1 = (const float*)d_in[6];
  const float* b1 = (const float*)d_in[7];
  const float* bn2g = (const float*)d_in[8];
  const float* bn2b = (const float*)d_in[9];
  const float* w2 = (const float*)d_in[10];
  const float* b2 = (const float*)d_in[11];
  float* out = (float*)d_out;
  float* ws = (float*)d_ws;

  constexpr size_t NF = (size_t)BB * CC * MM;
  float* featb = ws;  // feat, then t (in place)
  float* yb = featb + NF;
  float* h1b = yb + NF;
  float* gramp = h1b + NF;  // B*NT*G*32*32 partial Grams
  float* rsump = gramp + (size_t)BB * NT * GROUPS * DGRP * DGRP;
  float* s_sc = rsump + (size_t)BB * NT * CC;
  float* tsum = s_sc + (size_t)BB * CC;
  float* tsq = tsum + (size_t)BB * CC;
  float* i1sum = tsq + (size_t)BB * CC;
  float* i1sq = i1sum + (size_t)BB * CC;
  float* scA = i1sq + (size_t)BB * CC;
  float* shA = scA + (size_t)BB * CC;
  float* i2sum = shA + (size_t)BB * CC;
  float* i2sq = i2sum + (size_t)BB * NT * CC;
  float* scB = i2sq + (size_t)BB * NT * CC;
  float* shB = scB + (size_t)BB * CC;

  const dim3 gg(NT, BB);

  // 1) feat = Wl @ x  (+ Gram/rowsum partials)
  gemm128<1><<<gg, 256, 0, stream>>>(x, wl, nullptr, nullptr, nullptr, featb,
                                     nullptr, gramp, rsump, nullptr, nullptr);
  // 2) covariance -> Newton-Schulz sqrtm -> channel scales s
  ns_sqrtm_kernel<<<BB * GROUPS, 256, 0, stream>>>(gramp, rsump, s_sc);
  // 3) t = s*feat + x ; group-norm partial stats
  t_gn_kernel<<<BB * CC, 256, 0, stream>>>(x, featb, s_sc, tsum, tsq);
  // 4) y = group_norm(t) ; instance-norm stats of y
  gn_apply_kernel<<<BB * CC, 256, 0, stream>>>(featb, tsum, tsq, gng, gnb, yb,
                                               i1sum, i1sq);
  // 5) collapse IN+BN+relu into per-(b,c) affine
  coef_kernel<<<1, 128, 0, stream>>>(i1sum, i1sq, 1, bn1g, bn1b, scA, shA);
  // 6) h1 = W1 @ relu(scA*y+shA) + b1  (+ stats partials)
  gemm128<2><<<gg, 256, 0, stream>>>(yb, w1, b1, scA, shA, h1b, nullptr,
                                     nullptr, nullptr, i2sum, i2sq);
  // 7) second affine collapse
  coef_kernel<<<1, 128, 0, stream>>>(i2sum, i2sq, NT, bn2g, bn2b, scB, shB);
  // 8) out = W2 @ relu(scB*h1+shB) + b2 + y
  gemm128<3><<<gg, 256, 0, stream>>>(h1b, w2, b2, scB, shB, out, yb, nullptr,
                                     nullptr, nullptr, nullptr);
}